// PointWiseConvolutionBatch_88175678587633
// MI455X (gfx1250) — compile-verified
//
#include <hip/hip_runtime.h>

#define NPTS  2048
#define CPB   128     // centers per block = blockDim.x (4 wave32 waves)
#define NOUT  9
#define QC    16      // kernel cells

typedef __attribute__((ext_vector_type(2))) float v2f;
typedef __attribute__((ext_vector_type(8))) float v8f;

__global__ __launch_bounds__(CPB) void pwconv_kernel(
    const float* __restrict__ pts,    // [B, NPTS, 3]
    const float* __restrict__ attrs,  // [B, NPTS, 1]
    const float* __restrict__ w,      // [NOUT, 1, QC]
    const float* __restrict__ bias,   // [NOUT]
    float* __restrict__ out)          // [B, NPTS, NOUT]
{
    __shared__ float px[NPTS], py[NPTS], pz[NPTS], at[NPTS];
    __shared__ float gridl[CPB][QC + 1];   // +1 pad: stride 17 coprime with 64 banks
    __shared__ float wpad[16][QC + 1];     // zero-padded 16x16 weight tile

    const int tid   = threadIdx.x;
    const int b     = blockIdx.y;
    const int cbase = blockIdx.x * CPB;

    const float* P = pts   + (size_t)b * NPTS * 3;
    const float* A = attrs + (size_t)b * NPTS;

    // Stage the whole batch in LDS (32 KB)
    for (int j = tid; j < NPTS; j += CPB) {
        px[j] = P[3 * j + 0];
        py[j] = P[3 * j + 1];
        pz[j] = P[3 * j + 2];
        at[j] = A[j];
    }
    // Stage zero-padded weights (unconditional writes -> no exec churn later)
#pragma unroll
    for (int idx = tid; idx < 256; idx += CPB) {
        const int o = idx >> 4;
        const int q = idx & 15;
        wpad[o][q] = (o < NOUT) ? w[o * QC + q] : 0.0f;
    }
    __syncthreads();

    // ---- Stage 1: pairwise binning; one lane == one center ----
    const int   ci = cbase + tid;
    const float xi = px[ci], yi = py[ci], zi = pz[ci];

    float s[QC];
    float c[QC];
#pragma unroll
    for (int q = 0; q < QC; ++q) { s[q] = 0.0f; c[q] = 0.0f; }

    for (int j = 0; j < NPTS; ++j) {
        const float dx = xi - px[j];
        const float dy = yi - py[j];
        const float dz = zi - pz[j];
        float d2 = dx * dx;
        d2 = fmaf(dy, dy, d2);
        d2 = fmaf(dz, dz, d2);
        const float dist = sqrtf(d2);

        const int oct = (dx >= 0.0f ? 1 : 0) | (dy >= 0.0f ? 2 : 0) | (dz >= 0.0f ? 4 : 0);
        int cell = (dist < 0.05f ? 0 : 8) + oct;      // shell*8 + octant
        cell = (dist <= 0.1f) ? cell : QC;            // QC == invalid bucket

        const float aj = at[j];
#pragma unroll
        for (int q = 0; q < QC; ++q) {
            const bool hit = (cell == q);
            s[q] += hit ? aj   : 0.0f;
            c[q] += hit ? 1.0f : 0.0f;
        }
    }

    // ---- Stage 2: per-cell mean -> LDS grid tile ----
#pragma unroll
    for (int q = 0; q < QC; ++q)
        gridl[tid][q] = s[q] / fmaxf(c[q], 1.0f);
    __syncthreads();

    // ---- Stage 3: [16 centers x 16 cells] x [16 cells x 16(9) outs] via WMMA f32 ----
    const int lane = tid & 31;
    const int wv   = tid >> 5;     // wave id 0..3
    const int m    = lane & 15;    // M (centers) for A, N (out-ch) for B
    const int h    = lane >> 4;    // half-wave

#pragma unroll
    for (int t = 0; t < 2; ++t) {
        const int c0 = wv * 32 + t * 16;   // local center base of this 16-tile
        v8f acc = {};
#pragma unroll
        for (int kk = 0; kk < 4; ++kk) {
            const int k0 = kk * 4 + h * 2;   // K pair handled by this half-wave
            v2f av, bv;
            // A (16x4 f32): lane = M + 16*(K/2), vgpr = K%2
            av.x = gridl[c0 + m][k0 + 0];
            av.y = gridl[c0 + m][k0 + 1];
            // B (4x16 f32): lane = N + 16*(K/2), vgpr = K%2 ; zero-padded rows N >= 9
            bv.x = wpad[m][k0 + 0];
            bv.y = wpad[m][k0 + 1];
            acc = __builtin_amdgcn_wmma_f32_16x16x4_f32(
                false, av, false, bv, (short)0, acc, false, false);
        }
        // D (16x16 f32): vgpr r -> M = r + 8*h ; N = lane%16
        if (m < NOUT) {
            const float bo = bias[m];
#pragma unroll
            for (int r = 0; r < 8; ++r) {
                const int center = cbase + c0 + r + 8 * h;
                out[((size_t)b * NPTS + center) * NOUT + m] = acc[r] + bo;
            }
        }
    }
}

extern "C" void kernel_launch(void* const* d_in, const int* in_sizes, int n_in,
                              void* d_out, int out_size, void* d_ws, size_t ws_size,
                              hipStream_t stream) {
    const float* pts   = (const float*)d_in[0];   // [B,2048,3]
    const float* attrs = (const float*)d_in[1];   // [B,2048,1]
    const float* w     = (const float*)d_in[2];   // [9,1,16]
    const float* bias  = (const float*)d_in[3];   // [9]
    float* out = (float*)d_out;                   // [B,2048,9]

    const int B = in_sizes[0] / (NPTS * 3);
    dim3 grid(NPTS / CPB, B);
    pwconv_kernel<<<grid, CPB, 0, stream>>>(pts, attrs, w, bias, out);
}